// FRCNNModel_1967095022074
// MI455X (gfx1250) — compile-verified
//
#include <hip/hip_runtime.h>
#include <stdint.h>

// ---------------- constants ----------------
#define K_TOP    4096
#define K_WORDS  128          // 4096 / 32
#define CAND_CAP 8192
#define IMG_W_F  1333.0f
#define IMG_H_F  800.0f
#define BBOX_CLIP_F 4.135166556742356f   // log(1000/16)
#define NMS_T    0.7f

// workspace layout (bytes)
#define OFF_HIST   0u           // 256 u32 radix histogram
#define OFF_STATE  1024u        // [0]=prefix, [1]=K remaining, [2]=cand count
#define OFF_CAND   2048u        // 8192 u64 candidate keys (64 KB)
#define OFF_TOPIDX 67584u       // 4096 u32
#define OFF_TOPSC  83968u       // 4096 f32
#define OFF_TOPBOX 100352u      // 4096*4 f32
#define OFF_MASK   165888u      // 4096*128 u32 suppression bits (2 MB)
#define OFF_REM    2263040u     // 128 u32 final removed-mask

typedef float v2f  __attribute__((ext_vector_type(2)));
typedef float v8f  __attribute__((ext_vector_type(8)));
typedef unsigned int u32x4 __attribute__((ext_vector_type(4)));
typedef int   i32x4 __attribute__((ext_vector_type(4)));
typedef int   i32x8 __attribute__((ext_vector_type(8)));

__device__ __forceinline__ unsigned ordf(float f) {
  unsigned u = __float_as_uint(f);
  return (u & 0x80000000u) ? ~u : (u | 0x80000000u);   // order-preserving map
}

// ---- Tensor Data Mover: 1-D tile of f32 from global -> LDS (per-wave DMA) ----
__device__ __forceinline__ void tdm_load_f32(unsigned ldsByteOff, const void* gptr,
                                             unsigned nFloats) {
  unsigned long long ga = (unsigned long long)(uintptr_t)gptr;
  u32x4 g0;
  g0[0] = 1u;                                               // count=1, user mode
  g0[1] = ldsByteOff;                                       // lds_addr
  g0[2] = (unsigned)(ga & 0xFFFFFFFFull);                   // global_addr[31:0]
  g0[3] = (unsigned)((ga >> 32) & 0x01FFFFFFull) | (2u << 30); // addr[56:32] | type=2
  i32x8 g1;
  g1[0] = (int)(2u << 16);                                  // data_size = 4B
  g1[1] = (int)((nFloats & 0xFFFFu) << 16);                 // tensor_dim0 lo16
  g1[2] = (int)((nFloats >> 16) | (1u << 16));              // tensor_dim0 hi | tensor_dim1=1
  g1[3] = (int)((nFloats & 0xFFFFu) << 16);                 // tile_dim0 = nFloats
  g1[4] = 1;                                                // tile_dim1=1, tile_dim2=0
  g1[5] = (int)nFloats;                                     // tensor_dim0_stride lo32
  g1[6] = 0;
  g1[7] = 0;
  i32x4 z4 = (i32x4){0,0,0,0};
#if defined(__clang_major__) && (__clang_major__ >= 23)
  i32x8 z8 = (i32x8){0,0,0,0,0,0,0,0};
  __builtin_amdgcn_tensor_load_to_lds(g0, g1, z4, z4, z8, 0);
#else
  __builtin_amdgcn_tensor_load_to_lds(g0, g1, z4, z4, 0);
#endif
}

// ---------------- 1) init per-call state ----------------
__global__ void rpn_init(unsigned* hist, unsigned* state) {
  int t = threadIdx.x;
  hist[t] = 0u;
  if (t == 0) { state[0] = 0u; state[1] = (unsigned)K_TOP; state[2] = 0u; }
}

// ---------------- 2) radix-select histogram pass ----------------
__global__ void rpn_hist(const float* __restrict__ scores, int n,
                         unsigned* __restrict__ hist,
                         const unsigned* __restrict__ state, int p) {
  __shared__ unsigned lh[256];
  lh[threadIdx.x] = 0u;
  __syncthreads();
  int i = blockIdx.x * blockDim.x + threadIdx.x;
  if (i < n) {
    unsigned u = ordf(scores[i]);
    unsigned pfx = state[0];
    bool match = (p == 0) || ((u >> (32 - 8 * p)) == pfx);
    if (match) atomicAdd(&lh[(u >> (24 - 8 * p)) & 0xFFu], 1u);
  }
  __syncthreads();
  unsigned v = lh[threadIdx.x];
  if (v) atomicAdd(&hist[threadIdx.x], v);
}

// ---------------- 3) pick radix digit, update prefix ----------------
__global__ void rpn_select(unsigned* hist, unsigned* state) {
  if (threadIdx.x != 0) return;
  unsigned K = state[1];
  unsigned acc = 0; int chosen = 0;
  for (int b = 255; b >= 0; --b) {
    acc += hist[b];
    if (acc >= K) { chosen = b; break; }
  }
  unsigned above = acc - hist[chosen];       // strictly greater digits
  state[1] = K - above;
  state[0] = (state[0] << 8) | (unsigned)chosen;
  for (int b = 0; b < 256; ++b) hist[b] = 0u;
}

// ---------------- 4) compact candidates >= 24-bit prefix ----------------
__global__ void rpn_compact(const float* __restrict__ scores, int n,
                            unsigned* __restrict__ state,
                            unsigned long long* __restrict__ cand) {
  int i = blockIdx.x * blockDim.x + threadIdx.x;
  if (i >= n) return;
  unsigned u = ordf(scores[i]);
  if ((u >> 8) >= state[0]) {
    unsigned pos = atomicAdd(&state[2], 1u);
    if (pos < CAND_CAP)
      cand[pos] = ((unsigned long long)u << 32) | (0xFFFFFFFFu - (unsigned)i);
  }
}

// ---------------- 5) bitonic sort 8192 keys in LDS, emit top-4096 indices ----
__global__ __launch_bounds__(1024) void rpn_sort(const unsigned long long* __restrict__ cand,
                                                 const unsigned* __restrict__ state,
                                                 unsigned* __restrict__ topIdx) {
  __shared__ unsigned long long keys[CAND_CAP];   // 64 KB LDS
  int t = threadIdx.x;
  unsigned cnt = state[2];
  if (cnt > CAND_CAP) cnt = CAND_CAP;
  for (int i = t; i < CAND_CAP; i += 1024)
    keys[i] = (i < (int)cnt) ? cand[i] : 0ull;
  __syncthreads();
  for (int k = 2; k <= CAND_CAP; k <<= 1) {
    for (int j = k >> 1; j > 0; j >>= 1) {
      for (int i = t; i < CAND_CAP; i += 1024) {
        int ixj = i ^ j;
        if (ixj > i) {
          unsigned long long x = keys[i], y = keys[ixj];
          bool up = (i & k) == 0;                 // descending sort
          if (up ? (x < y) : (x > y)) { keys[i] = y; keys[ixj] = x; }
        }
      }
      __syncthreads();
    }
  }
  for (int i = t; i < K_TOP; i += 1024)
    topIdx[i] = 0xFFFFFFFFu - (unsigned)(keys[i] & 0xFFFFFFFFull);
}

// ---------------- 6) decode+clip only the 4096 winners ----------------
__global__ void rpn_decode(const float* __restrict__ scores,
                           const float* __restrict__ anchors,
                           const float* __restrict__ deltas,
                           const unsigned* __restrict__ topIdx,
                           float* __restrict__ topBoxes,
                           float* __restrict__ topScores, int n) {
  int k = blockIdx.x * blockDim.x + threadIdx.x;
  if (k >= K_TOP) return;
  unsigned idx = topIdx[k];
  if (idx >= (unsigned)n) {                 // defensive (cannot happen for N>=4096)
    topBoxes[k*4+0] = topBoxes[k*4+1] = topBoxes[k*4+2] = topBoxes[k*4+3] = 0.f;
    topScores[k] = -1.0f;
    return;
  }
  const float* a = anchors + (size_t)idx * 4;
  const float* d = deltas  + (size_t)idx * 4;
  float w  = a[2] - a[0], h = a[3] - a[1];
  float cx = a[0] + 0.5f * w, cy = a[1] + 0.5f * h;
  float dx = d[0] / 10.0f, dy = d[1] / 10.0f;
  float dw = fminf(d[2] / 5.0f, BBOX_CLIP_F);
  float dh = fminf(d[3] / 5.0f, BBOX_CLIP_F);
  float pcx = dx * w + cx, pcy = dy * h + cy;
  float pw = expf(dw) * w, ph = expf(dh) * h;
  float x1 = pcx - 0.5f * pw, y1 = pcy - 0.5f * ph;
  float x2 = pcx + 0.5f * pw, y2 = pcy + 0.5f * ph;
  topBoxes[k*4+0] = fminf(fmaxf(x1, 0.f), IMG_W_F);
  topBoxes[k*4+1] = fminf(fmaxf(y1, 0.f), IMG_H_F);
  topBoxes[k*4+2] = fminf(fmaxf(x2, 0.f), IMG_W_F);
  topBoxes[k*4+3] = fminf(fmaxf(y2, 0.f), IMG_H_F);
  topScores[k] = scores[idx];
}

// ---------------- 7) IoU suppression bitmask: TDM tiles + WMMA area-sum -----
// One wave per 16-row x 32-col tile. S[m,n] = area_m + area_n computed as a
// genuine 16x16x4 f32 matmul:  A[m,:]=[area_m,1,1,0], B[:,n]=[1,area_n,0,0]
// (coefficients chosen to be invariant under the two possible K-orderings of
// the f32 B-matrix VGPR layout).
__global__ __launch_bounds__(32) void rpn_iou(const float* __restrict__ topBoxes,
                                              unsigned* __restrict__ mask) {
  __shared__ __align__(16) float sb[(16 + 32) * 4];  // rows @0..256B, cols @256..768B
  const int lane = threadIdx.x;
  const int r0 = blockIdx.y * 16;
  const int c0 = blockIdx.x * 32;

  // DMA the row tile (64 f32) and col tile (128 f32) into LDS via the TDM.
  tdm_load_f32(0u,   topBoxes + (size_t)r0 * 4, 64u);
  tdm_load_f32(256u, topBoxes + (size_t)c0 * 4, 128u);
  __builtin_amdgcn_s_wait_tensorcnt(0);
  // Escape the LDS pointer into the asm + memory clobber: the TDM write is
  // invisible to the compiler, so without this it treats sb[] as undef and
  // const-folds the whole tile computation away (observed in round 1 asm).
  const float* RB = sb;           // 16 row boxes
  const float* CB = sb + 64;      // 32 col boxes
  __asm volatile("" : : "v"(RB), "v"(CB) : "memory");

  const int q = lane & 15;
  const bool hi = lane >= 16;

  // Column boxes owned by this lane (subtile A: cols c0+q, subtile B: +16).
  float ax1 = CB[q*4+0], ay1 = CB[q*4+1], ax2 = CB[q*4+2], ay2 = CB[q*4+3];
  float bx1 = CB[(q+16)*4+0], by1 = CB[(q+16)*4+1];
  float bx2 = CB[(q+16)*4+2], by2 = CB[(q+16)*4+3];
  float areaColA = (ax2 - ax1) * (ay2 - ay1);
  float areaColB = (bx2 - bx1) * (by2 - by1);
  float areaRow  = (RB[q*4+2] - RB[q*4+0]) * (RB[q*4+3] - RB[q*4+1]);

  // A-matrix 16x4 layout: lanes0-15 hold K0(vgpr0)/K1(vgpr1), lanes16-31 K2/K3.
  v2f A  = { hi ? 1.0f : areaRow, hi ? 0.0f : 1.0f };
  v2f B0 = { hi ? 0.0f : 1.0f,    hi ? 0.0f : areaColA };
  v2f B1 = { hi ? 0.0f : 1.0f,    hi ? 0.0f : areaColB };
  v8f Z  = { 0.f, 0.f, 0.f, 0.f, 0.f, 0.f, 0.f, 0.f };

  v8f S0 = __builtin_amdgcn_wmma_f32_16x16x4_f32(false, A, false, B0, (short)0, Z, false, false);
  v8f S1 = __builtin_amdgcn_wmma_f32_16x16x4_f32(false, A, false, B1, (short)0, Z, false, false);

  unsigned myword = 0u;
#pragma unroll
  for (int r = 0; r < 8; ++r) {
    int m = r + (hi ? 8 : 0);      // C/D layout: vgpr r = rows r (lo half) / r+8 (hi half)
    float rx1 = RB[m*4+0], ry1 = RB[m*4+1], rx2 = RB[m*4+2], ry2 = RB[m*4+3];

    float iwA = fminf(rx2, ax2) - fmaxf(rx1, ax1);
    float ihA = fminf(ry2, ay2) - fmaxf(ry1, ay1);
    float interA = fmaxf(iwA, 0.f) * fmaxf(ihA, 0.f);
    bool cA = (interA / (S0[r] - interA + 1e-9f)) > NMS_T;

    float iwB = fminf(rx2, bx2) - fmaxf(rx1, bx1);
    float ihB = fminf(ry2, by2) - fmaxf(ry1, by1);
    float interB = fmaxf(iwB, 0.f) * fmaxf(ihB, 0.f);
    bool cB = (interB / (S1[r] - interB + 1e-9f)) > NMS_T;

    unsigned balA = __builtin_amdgcn_ballot_w32(cA);
    unsigned balB = __builtin_amdgcn_ballot_w32(cB);
    unsigned wl = (balA & 0xFFFFu) | (balB << 16);          // row r,  cols 0..31
    unsigned wh = (balA >> 16) | (balB & 0xFFFF0000u);      // row r+8, cols 0..31
    if (lane == r)     myword = wl;
    if (lane == r + 8) myword = wh;
  }
  if (lane < 16)
    mask[(size_t)(r0 + lane) * K_WORDS + blockIdx.x] = myword;
}

// ---------------- 8) sequential greedy-NMS scan (1 block) ----------------
__global__ __launch_bounds__(128) void rpn_nms(const unsigned* __restrict__ mask,
                                               unsigned* __restrict__ remOut) {
  __shared__ unsigned rem[K_WORDS];
  int t = threadIdx.x;
  rem[t] = 0u;
  __syncthreads();
  for (int i = 0; i < K_TOP; ++i) {
    unsigned wi = (unsigned)i >> 5, bi = (unsigned)i & 31u;
    bool kept = ((rem[wi] >> bi) & 1u) == 0u;
    __syncthreads();
    if (kept) {
      unsigned m = mask[(size_t)i * K_WORDS + t];
      if ((unsigned)t == wi) m &= ~((2u << bi) - 1u);   // only suppress j > i
      if ((unsigned)t >= wi) rem[t] |= m;
    }
    __syncthreads();
  }
  remOut[t] = rem[t];
}

// ---------------- 9) emit [4096,5] output ----------------
__global__ void rpn_out(const float* __restrict__ topBoxes,
                        const float* __restrict__ topScores,
                        const unsigned* __restrict__ rem,
                        float* __restrict__ out) {
  int k = blockIdx.x * blockDim.x + threadIdx.x;
  if (k >= K_TOP) return;
  bool kept = ((rem[k >> 5] >> (k & 31)) & 1u) == 0u;
  float s = kept ? topScores[k] : 0.0f;
  out[k*5+0] = kept ? topBoxes[k*4+0] : 0.0f;
  out[k*5+1] = kept ? topBoxes[k*4+1] : 0.0f;
  out[k*5+2] = kept ? topBoxes[k*4+2] : 0.0f;
  out[k*5+3] = kept ? topBoxes[k*4+3] : 0.0f;
  out[k*5+4] = s;
}

// ---------------- host-side launch ----------------
extern "C" void kernel_launch(void* const* d_in, const int* in_sizes, int n_in,
                              void* d_out, int out_size, void* d_ws, size_t ws_size,
                              hipStream_t stream) {
  const float* scores  = (const float*)d_in[0];
  const float* anchors = (const float*)d_in[1];
  const float* deltas  = (const float*)d_in[2];
  float* out = (float*)d_out;
  int N = in_sizes[0];

  char* ws = (char*)d_ws;
  unsigned*           hist   = (unsigned*)(ws + OFF_HIST);
  unsigned*           state  = (unsigned*)(ws + OFF_STATE);
  unsigned long long* cand   = (unsigned long long*)(ws + OFF_CAND);
  unsigned*           topIdx = (unsigned*)(ws + OFF_TOPIDX);
  float*              topSc  = (float*)(ws + OFF_TOPSC);
  float*              topBox = (float*)(ws + OFF_TOPBOX);
  unsigned*           maskp  = (unsigned*)(ws + OFF_MASK);
  unsigned*           remp   = (unsigned*)(ws + OFF_REM);

  int nb = (N + 255) / 256;

  rpn_init<<<1, 256, 0, stream>>>(hist, state);
  // 3x 8-bit radix-select passes -> 24-bit prefix of the 4096th largest score
  rpn_hist<<<nb, 256, 0, stream>>>(scores, N, hist, state, 0);
  rpn_select<<<1, 32, 0, stream>>>(hist, state);
  rpn_hist<<<nb, 256, 0, stream>>>(scores, N, hist, state, 1);
  rpn_select<<<1, 32, 0, stream>>>(hist, state);
  rpn_hist<<<nb, 256, 0, stream>>>(scores, N, hist, state, 2);
  rpn_select<<<1, 32, 0, stream>>>(hist, state);

  rpn_compact<<<nb, 256, 0, stream>>>(scores, N, state, cand);
  rpn_sort<<<1, 1024, 0, stream>>>(cand, state, topIdx);
  rpn_decode<<<K_TOP / 256, 256, 0, stream>>>(scores, anchors, deltas, topIdx,
                                              topBox, topSc, N);
  rpn_iou<<<dim3(K_WORDS, K_TOP / 16), 32, 0, stream>>>(topBox, maskp);
  rpn_nms<<<1, 128, 0, stream>>>(maskp, remp);
  rpn_out<<<K_TOP / 256, 256, 0, stream>>>(topBox, topSc, remp, out);
}